// Model_39676907884816
// MI455X (gfx1250) — compile-verified
//
#include <hip/hip_runtime.h>
#include <math.h>

typedef __attribute__((ext_vector_type(2))) float v2f;
typedef __attribute__((ext_vector_type(4))) float v4f;
typedef __attribute__((ext_vector_type(8))) float v8f;

#define N_Q      8
#define N_KV     16
#define N_ROWS   24
#define D_DIM    4194304
#define TILE_K   256
#define TILE_KP  260          // padded LDS row stride (floats) to spread banks
#define NTHREADS 256
#define NWAVES   8
#define MAX_NB   1024
#define NOUT     136          // 128 qk partials + 8 norm partials

// ---------------------------------------------------------------------------
// Kernel 1: per-block partial Gram reduction using V_WMMA_F32_16X16X4_F32.
//   A (16x4 f32): rows 0..7 = q rows, rows 8..15 duplicate q (discarded).
//   B (4x16 f32): columns = kv rows.  C accumulates qk partials over this
//   block's k-slices.  q-norms accumulated with plain VALU FMAs.
// Deterministic: each block writes its own slot in part[NOUT][nb].
// ---------------------------------------------------------------------------
__global__ __launch_bounds__(NTHREADS)
void qk_partial_kernel(const float* __restrict__ qkv, float* __restrict__ part)
{
    __shared__ float tile[N_ROWS * TILE_KP];     // 24*260*4  = 24960 B
    __shared__ float red[NWAVES * 256];          // 8 waves x 256 C entries
    __shared__ float rnorm[NWAVES * 32];         // 8 waves x 32 lanes

    const int t    = threadIdx.x;
    const int lane = t & 31;
    const int wave = t >> 5;

    v8f  c    = {};          // 16x16 f32 accumulator (8 VGPRs, wave-striped)
    float nacc = 0.0f;

    const int numTiles = D_DIM / TILE_K;         // 16384
    for (int tileId = blockIdx.x; tileId < numTiles; tileId += gridDim.x) {
        const int c0 = tileId * TILE_K;

        // ---- cooperative tile load: 24 rows x 256 floats = 1536 float4 ----
        #pragma unroll
        for (int i = 0; i < 6; ++i) {
            const int idx = t + i * NTHREADS;    // 0..1535
            const int row = idx >> 6;            // 64 float4 per row
            const int k4  = idx & 63;
            const float* src = qkv + (size_t)row * D_DIM + c0 + k4 * 4;
            v4f v = *(const v4f*)src;
            *(v4f*)&tile[row * TILE_KP + k4 * 4] = v;
            // prefetch next tile this block will touch (1 MB ahead -> L2)
            __builtin_prefetch(src + (size_t)gridDim.x * TILE_K, 0, 1);
        }
        __syncthreads();

        // ---- WMMA over this wave's 32-k slice ----
        const int kbase = wave * 32;
        const int koff  = (lane >> 4) << 1;      // lanes 0-15 -> k{0,1}; 16-31 -> k{2,3}
        const int mrow  = lane & 15;
        const float* arow = &tile[(mrow & 7) * TILE_KP];          // q row (dup 8..15)
        const float* brow = &tile[(N_Q + mrow) * TILE_KP];        // kv row = B column
        #pragma unroll
        for (int kk = 0; kk < 32; kk += 4) {
            v2f a, b;
            a.x = arow[kbase + kk + koff];
            a.y = arow[kbase + kk + koff + 1];
            b.x = brow[kbase + kk + koff];
            b.y = brow[kbase + kk + koff + 1];
            c = __builtin_amdgcn_wmma_f32_16x16x4_f32(
                    false, a, false, b, (short)0, c, false, false);
        }

        // ---- q self-dot partials: lane covers q row (lane>>2), phase lane&3 ----
        {
            const int   qi = lane >> 2;
            const int   ks = lane & 3;
            const float* qp = &tile[qi * TILE_KP + kbase + ks];
            #pragma unroll
            for (int kk = 0; kk < 32; kk += 4) {
                const float x = qp[kk];
                nacc = fmaf(x, x, nacc);
            }
        }
        __syncthreads();
    }

    // ---- block reduction across the 8 waves ----
    #pragma unroll
    for (int v = 0; v < 8; ++v)
        red[wave * 256 + v * 32 + lane] = c[v];
    rnorm[wave * 32 + lane] = nacc;
    __syncthreads();

    {   // each thread owns one of the 256 C-fragment entries
        float s = 0.0f;
        #pragma unroll
        for (int w = 0; w < NWAVES; ++w) s += red[w * 256 + t];
        const int v = t >> 5, l = t & 31;
        const int M = v + ((l >> 4) << 3);       // VGPR layout: M = v (+8 for hi half)
        const int N = l & 15;
        if (M < N_Q)
            part[(size_t)(M * 16 + N) * gridDim.x + blockIdx.x] = s;
    }
    if (t < N_Q) {                               // norm rows
        float s = 0.0f;
        #pragma unroll
        for (int w = 0; w < NWAVES; ++w)
            #pragma unroll
            for (int ss = 0; ss < 4; ++ss)
                s += rnorm[w * 32 + t * 4 + ss];
        part[(size_t)(128 + t) * gridDim.x + blockIdx.x] = s;
    }
}

// ---------------------------------------------------------------------------
// Kernel 2: deterministic cross-block sum + scale + softmax + dropout.
// ---------------------------------------------------------------------------
__device__ __forceinline__ unsigned rotl32(unsigned v, int n) {
    return (v << n) | (v >> (32 - n));
}

// threefry2x32, 16 rounds, key = (0, 42), ctr = (0, idx)  (dropout mask source)
__device__ __forceinline__ unsigned threefry42(unsigned idx) {
    const unsigned k0 = 0u, k1 = 42u, k2 = k0 ^ k1 ^ 0x1BD11BDAu;
    unsigned x0 = 0u, x1 = idx;
    const int R0[4] = {13, 15, 26, 6};
    const int R1[4] = {17, 29, 16, 24};
    x0 += k0; x1 += k1;
    #pragma unroll
    for (int r = 0; r < 4; ++r) { x0 += x1; x1 = rotl32(x1, R0[r]); x1 ^= x0; }
    x0 += k1; x1 += k2 + 1u;
    #pragma unroll
    for (int r = 0; r < 4; ++r) { x0 += x1; x1 = rotl32(x1, R1[r]); x1 ^= x0; }
    x0 += k2; x1 += k0 + 2u;
    #pragma unroll
    for (int r = 0; r < 4; ++r) { x0 += x1; x1 = rotl32(x1, R0[r]); x1 ^= x0; }
    x0 += k0; x1 += k1 + 3u;
    #pragma unroll
    for (int r = 0; r < 4; ++r) { x0 += x1; x1 = rotl32(x1, R1[r]); x1 ^= x0; }
    x0 += k1; x1 += k2 + 4u;
    return x0 ^ x1;
}

__global__ __launch_bounds__(128)
void finalize_kernel(const float* __restrict__ part, float* __restrict__ out, int nb)
{
    __shared__ float qk[N_Q][N_KV];
    __shared__ float nrm[N_Q];

    const int t = threadIdx.x;                   // 0..127
    {
        float s = 0.0f;
        const float* p = part + (size_t)t * nb;
        for (int b = 0; b < nb; ++b) s += p[b];
        qk[t >> 4][t & 15] = s;
    }
    if (t < N_Q) {
        float s = 0.0f;
        const float* p = part + (size_t)(128 + t) * nb;
        for (int b = 0; b < nb; ++b) s += p[b];
        nrm[t] = s;
    }
    __syncthreads();

    const int i = t >> 4, j = t & 15;
    const float sc = rsqrtf(nrm[i] + 1e-12f);
    const float x  = qk[i][j] * sc;

    float m = -INFINITY;
    #pragma unroll
    for (int jj = 0; jj < N_KV; ++jj) m = fmaxf(m, qk[i][jj] * sc);
    float sum = 0.0f;
    #pragma unroll
    for (int jj = 0; jj < N_KV; ++jj) sum += __expf(qk[i][jj] * sc - m);
    const float p = __expf(x - m) / sum;

    const unsigned idx = (unsigned)(i * 16 + j);
    const float u = (float)(threefry42(idx) >> 8) * (1.0f / 16777216.0f);
    out[idx] = (u < 0.9f) ? p * (1.0f / 0.9f) : 0.0f;
}

// ---------------------------------------------------------------------------
extern "C" void kernel_launch(void* const* d_in, const int* in_sizes, int n_in,
                              void* d_out, int out_size, void* d_ws, size_t ws_size,
                              hipStream_t stream)
{
    const float* qkv  = (const float*)d_in[0];
    float*       out  = (float*)d_out;
    float*       part = (float*)d_ws;

    int nb = MAX_NB;
    const size_t need = (size_t)NOUT * sizeof(float);
    if (ws_size < (size_t)nb * need) {
        size_t fit = ws_size / need;
        nb = (fit < 1) ? 1 : (int)fit;
        if (nb > MAX_NB) nb = MAX_NB;
    }

    qk_partial_kernel<<<nb, NTHREADS, 0, stream>>>(qkv, part);
    finalize_kernel<<<1, 128, 0, stream>>>(part, out, nb);
}